// RoutingCapsules_11982958756266
// MI455X (gfx1250) — compile-verified
//
#include <hip/hip_runtime.h>
#include <hip/hip_bf16.h>

// Problem constants (from reference):
//   B=32 batches, I=2048 input caps, K=16 in-dim, J=64 out caps, D=32 cap dim
//   N = J*D = 2048 output features per input capsule
#define CB   32
#define CI   2048
#define CK   16
#define CJ   64
#define CD   32
#define CN   2048           // J*D
#define BJD  65536          // B*J*D = 32*64*32
#define NSPLIT 16           // partial s-buffers to spread atomic contention

typedef __attribute__((ext_vector_type(2))) float v2f;
typedef __attribute__((ext_vector_type(8))) float v8f;

// -------------------------------------------------------------------------
// Zero the partial s accumulators (must be clean before each routing pass).
__global__ void caps_zero_kernel(float* __restrict__ p, int n) {
    int idx = blockIdx.x * blockDim.x + threadIdx.x;
    if (idx < n) p[idx] = 0.0f;
}

// -------------------------------------------------------------------------
// Fused routing pass. One workgroup per input capsule i.
//   mode 0: c = 1/64 (first iteration, b-logits are all zero)
//   mode 1: agreement with vprev, b = a        (fresh logits), softmax -> c
//   mode 2: agreement with vprev, b = bprev+a  (read logits),  softmax -> c
// Then s_partial += c * u_hat  (u_hat recomputed via f32 WMMA into LDS).
__global__ __launch_bounds__(256, 1)
void caps_pass_kernel(const float* __restrict__ x,     // [B, I, K]
                      const float* __restrict__ W,     // [I, J, D, K] = [I, N, K]
                      const float* __restrict__ vprev, // [B, J, D]
                      float* __restrict__ bcoef,       // [B, I, J] logits workspace
                      float* __restrict__ spart,       // [NSPLIT, B, N]
                      int mode)
{
    extern __shared__ float smem[];
    float* lds_u = smem;                 // [32][2048] u_hat tile  (256 KB)
    float* abuf  = smem + CB * CN;       // [32][64] agreement     (8 KB)
    float* cbuf  = abuf + CB * CJ;       // [32][64] routing coeff (8 KB)

    const int i     = blockIdx.x;        // input capsule
    const int tid   = threadIdx.x;
    const int lane  = tid & 31;
    const int wave  = tid >> 5;          // 0..7
    const int lhalf = lane >> 4;         // 0/1 (K-half per ISA A/B layout)
    const int lm    = lane & 15;         // M (or N) index within tile

    // ---- A fragments: x[b, i, k] for both 16-row batch halves, 4 k-steps.
    // V_WMMA_F32_16X16X4_F32 A layout: VGPR0 = K=ks*4+{0|2}, VGPR1 = +1 (lane halves).
    v2f afrag[2][4];
#pragma unroll
    for (int h = 0; h < 2; ++h) {
#pragma unroll
        for (int ks = 0; ks < 4; ++ks) {
            const int b = h * 16 + lm;
            const int k = ks * 4 + lhalf * 2;
            afrag[h][ks] = *(const v2f*)(x + (size_t)b * (CI * CK) + (size_t)i * CK + k);
        }
    }

    // ---- GEMM: u_hat[b, n] = sum_k x[b,i,k] * W[i,n,k]; tile -> LDS.
    // Each wave owns 16 of the 128 n-tiles, both batch halves (W read once).
    const float* Wi = W + (size_t)i * (CN * CK);
    for (int nt = wave; nt < CN / 16; nt += 8) {
        const int nbase = nt * 16;
        v8f acc0 = {};
        v8f acc1 = {};
        const float* pw = Wi + (size_t)(nbase + lm) * CK + lhalf * 2;
#pragma unroll
        for (int ks = 0; ks < 4; ++ks) {
            v2f bfrag = *(const v2f*)(pw + ks * 4);   // B layout mirrors A (K in halves)
            acc0 = __builtin_amdgcn_wmma_f32_16x16x4_f32(
                       false, afrag[0][ks], false, bfrag, (short)0, acc0, false, false);
            acc1 = __builtin_amdgcn_wmma_f32_16x16x4_f32(
                       false, afrag[1][ks], false, bfrag, (short)0, acc1, false, false);
        }
        // C layout: VGPR r -> M = r + lhalf*8, N = nbase + lm
#pragma unroll
        for (int r = 0; r < 8; ++r) {
            lds_u[(r + lhalf * 8) * CN + nbase + lm]        = acc0[r];
            lds_u[(16 + r + lhalf * 8) * CN + nbase + lm]   = acc1[r];
        }
    }
    __syncthreads();

    // ---- Routing coefficients.
    if (mode == 0) {
        for (int p = tid; p < CB * CJ; p += 256) cbuf[p] = 1.0f / 64.0f;
    } else {
        // agreement a[b,j] = sum_d u_hat[b, j*32+d] * v[b,j,d]
        for (int p = tid; p < CB * CJ; p += 256) {
            const int b = p >> 6, j = p & 63;
            const float* pu = &lds_u[b * CN + j * CD];
            const float* pv = vprev + (size_t)(b * CJ + j) * CD;
            float a = 0.0f;
#pragma unroll
            for (int d = 0; d < CD; ++d) a += pu[d] * pv[d];
            float* pb = bcoef + ((size_t)b * CI + i) * CJ + j;
            float bn = (mode == 2) ? (*pb + a) : a;
            *pb = bn;                       // keep logits for the next pass
            abuf[p] = bn;
        }
        __syncthreads();
        // softmax over j (64) per batch row; 32 threads, one row each
        if (tid < CB) {
            const int b = tid;
            float mx = -3.402823466e38f;
            for (int j = 0; j < CJ; ++j) mx = fmaxf(mx, abuf[b * CJ + j]);
            float se = 0.0f;
            for (int j = 0; j < CJ; ++j) {
                float e = __expf(abuf[b * CJ + j] - mx);
                cbuf[b * CJ + j] = e;
                se += e;
            }
            float inv = 1.0f / se;
            for (int j = 0; j < CJ; ++j) cbuf[b * CJ + j] *= inv;
        }
    }
    __syncthreads();

    // ---- Accumulate s_partial[b,n] += c[b, n>>5] * u_hat[b,n]
    float* sp = spart + (size_t)(blockIdx.x & (NSPLIT - 1)) * BJD;
    for (int e = tid; e < BJD; e += 256) {
        const int b = e >> 11;
        const int j = (e & 2047) >> 5;
        unsafeAtomicAdd(&sp[e], cbuf[b * CJ + j] * lds_u[e]);  // global_atomic_add_f32
    }
}

// -------------------------------------------------------------------------
// Fold NSPLIT partials and squash. One wave32 per (b,j) row, lane == d.
__global__ __launch_bounds__(256)
void caps_squash_kernel(const float* __restrict__ spart, float* __restrict__ v)
{
    const int lane = threadIdx.x & 31;
    const int row  = blockIdx.x * 8 + (threadIdx.x >> 5);   // (b*64 + j), 0..2047
    const int idx  = row * CD + lane;
    float t = 0.0f;
#pragma unroll
    for (int s = 0; s < NSPLIT; ++s) t += spart[(size_t)s * BJD + idx];
    float sq = t * t;
#pragma unroll
    for (int off = 16; off >= 1; off >>= 1) sq += __shfl_xor(sq, off, 32);
    const float scale = sq / (1.0f + sq);
    v[idx] = scale * t * rsqrtf(sq + 1e-8f);
}

// -------------------------------------------------------------------------
extern "C" void kernel_launch(void* const* d_in, const int* in_sizes, int n_in,
                              void* d_out, int out_size, void* d_ws, size_t ws_size,
                              hipStream_t stream) {
    const float* x = (const float*)d_in[0];   // [32, 2048, 16]
    const float* W = (const float*)d_in[1];   // [1, 2048, 64, 32, 16]
    float* out = (float*)d_out;               // [32, 64, 32]

    float* spart = (float*)d_ws;              // NSPLIT * BJD floats (4 MB)
    float* vbuf  = spart + (size_t)NSPLIT * BJD;   // BJD floats (256 KB)
    float* bcoef = vbuf + BJD;                // B*I*J floats (16 MB)

    const size_t shbytes = (size_t)(CB * CN + 2 * CB * CJ) * sizeof(float); // 272 KB
    const int nzero = NSPLIT * BJD;

    // it = 0: uniform routing coefficients
    caps_zero_kernel<<<nzero / 256, 256, 0, stream>>>(spart, nzero);
    caps_pass_kernel<<<CI, 256, shbytes, stream>>>(x, W, vbuf, bcoef, spart, 0);
    caps_squash_kernel<<<(CB * CJ) / 8, 256, 0, stream>>>(spart, vbuf);

    // it = 1: b = agreement(v0)
    caps_zero_kernel<<<nzero / 256, 256, 0, stream>>>(spart, nzero);
    caps_pass_kernel<<<CI, 256, shbytes, stream>>>(x, W, vbuf, bcoef, spart, 1);
    caps_squash_kernel<<<(CB * CJ) / 8, 256, 0, stream>>>(spart, vbuf);

    // it = 2: b += agreement(v1); final v -> d_out
    caps_zero_kernel<<<nzero / 256, 256, 0, stream>>>(spart, nzero);
    caps_pass_kernel<<<CI, 256, shbytes, stream>>>(x, W, vbuf, bcoef, spart, 2);
    caps_squash_kernel<<<(CB * CJ) / 8, 256, 0, stream>>>(spart, out);
}